// GIN_68959994905005
// MI455X (gfx1250) — compile-verified
//
#include <hip/hip_runtime.h>

#define GIN_D_IN 68
#define GIN_D_H  64
#define GIN_D3   192
#define GIN_BN_EPS 1e-5f

typedef float v2f __attribute__((ext_vector_type(2)));
typedef float v8f __attribute__((ext_vector_type(8)));

// ---------------------------------------------------------------------------
// agg init: strided row copy (cols multiple of 4)
// ---------------------------------------------------------------------------
__global__ void gin_copy_rows(float* __restrict__ dst, int dst_stride,
                              const float* __restrict__ src, int src_stride,
                              int rows, int cols4) {
  int tid = blockIdx.x * blockDim.x + threadIdx.x;
  int total = rows * cols4;
  if (tid >= total) return;
  int r  = tid / cols4;
  int c4 = tid - r * cols4;
  const float4* s = (const float4*)(src + (size_t)r * src_stride) + c4;
  float4*       d = (float4*)(dst + (size_t)r * dst_stride) + c4;
  *d = *s;
}

// ---------------------------------------------------------------------------
// edge scatter-add: agg[dst] += h[src]  (one thread per edge x float4 chunk)
// ---------------------------------------------------------------------------
__global__ void gin_scatter_add(float* __restrict__ agg, int agg_stride,
                                const float* __restrict__ h, int h_stride,
                                const long long* __restrict__ src_idx,
                                const long long* __restrict__ dst_idx,
                                int n_edges, int cols4) {
  int tid = blockIdx.x * blockDim.x + threadIdx.x;
  int total = n_edges * cols4;
  if (tid >= total) return;
  int e  = tid / cols4;
  int c4 = tid - e * cols4;
  long long s = src_idx[e];
  long long d = dst_idx[e];
  const float4 v = *((const float4*)(h + (size_t)s * h_stride) + c4);
  float* o = agg + (size_t)d * agg_stride + 4 * c4;
  atomicAdd(o + 0, v.x);
  atomicAdd(o + 1, v.y);
  atomicAdd(o + 2, v.z);
  atomicAdd(o + 3, v.w);
}

// ---------------------------------------------------------------------------
// GEMM (M x K) @ (K x 64) + bias [+ folded BN] + ReLU, fp32 WMMA 16x16x4.
// One wave computes a 16-row x 64-col tile with 4 accumulators.
// ---------------------------------------------------------------------------
__launch_bounds__(256)
__global__ void gin_mlp_gemm(const float* __restrict__ A, int lda, int K,
                             const float* __restrict__ W,     // K x 64 row-major
                             const float* __restrict__ bias,  // 64
                             const float* __restrict__ bn_g,  // 64 or null
                             const float* __restrict__ bn_be,
                             const float* __restrict__ bn_rm,
                             const float* __restrict__ bn_rv,
                             float* __restrict__ Out, int ldo,
                             int rows, int m_tiles) {
  int wave = (blockIdx.x * blockDim.x + threadIdx.x) >> 5;
  if (wave >= m_tiles) return;                 // wave-uniform: EXEC stays full
  int lane = threadIdx.x & 31;
  int half = lane >> 4;
  int l    = lane & 15;
  int row0 = wave * 16;

  int ra = row0 + l;
  if (ra > rows - 1) ra = rows - 1;            // clamp A reads (stores guarded)
  const float* __restrict__ arow = A + (size_t)ra * lda;

  v8f acc[4];
  acc[0] = v8f{}; acc[1] = v8f{}; acc[2] = v8f{}; acc[3] = v8f{};

  for (int k0 = 0; k0 < K; k0 += 4) {
    int ka = k0 + 2 * half;                    // A: vgpr v holds K = v + 2*half
    v2f a;
    a.x = arow[ka];
    a.y = arow[ka + 1];
    const float* wr0 = W + (size_t)ka * GIN_D_H + l;   // B row K=ka
    const float* wr1 = wr0 + GIN_D_H;                  // B row K=ka+1
#pragma unroll
    for (int t = 0; t < 4; ++t) {
      v2f b;
      b.x = wr0[t * 16];
      b.y = wr1[t * 16];
      acc[t] = __builtin_amdgcn_wmma_f32_16x16x4_f32(
          false, a, false, b, (short)0, acc[t], false, false);
    }
  }

#pragma unroll
  for (int t = 0; t < 4; ++t) {
    int col = t * 16 + l;
    float bi = bias[col];
    float scale = 1.0f, shift = 0.0f;
    if (bn_g) {                                // eval-mode BN folded to affine
      float g  = bn_g[col];
      float be = bn_be[col];
      float rm = bn_rm[col];
      float rv = bn_rv[col];
      scale = g * rsqrtf(rv + GIN_BN_EPS);
      shift = be - rm * scale;
    }
#pragma unroll
    for (int v = 0; v < 8; ++v) {
      int row = row0 + v + 8 * half;           // C/D: vgpr v -> M = v + 8*half
      float val = acc[t][v] + bi;
      if (bn_g) val = fmaf(val, scale, shift);
      val = fmaxf(val, 0.0f);
      if (row < rows) Out[(size_t)row * ldo + col] = val;
    }
  }
}

// ---------------------------------------------------------------------------
// Fused head: out = relu(H @ W1 + b1) @ w2 + b2, H is N x 192.
// One wave: 16 rows x full 192 cols (12 accumulators), then lane reduction.
// ---------------------------------------------------------------------------
__launch_bounds__(256)
__global__ void gin_final_head(const float* __restrict__ H,   // N x 192
                               const float* __restrict__ W1,  // 192 x 192
                               const float* __restrict__ b1,  // 192
                               const float* __restrict__ w2,  // 192
                               const float* __restrict__ b2,  // 1
                               float* __restrict__ out,
                               int rows, int m_tiles) {
  int wave = (blockIdx.x * blockDim.x + threadIdx.x) >> 5;
  if (wave >= m_tiles) return;
  int lane = threadIdx.x & 31;
  int half = lane >> 4;
  int l    = lane & 15;
  int row0 = wave * 16;

  int ra = row0 + l;
  if (ra > rows - 1) ra = rows - 1;
  const float* __restrict__ hrow = H + (size_t)ra * GIN_D3;

  v8f acc[12];
#pragma unroll
  for (int t = 0; t < 12; ++t) acc[t] = v8f{};

  for (int k0 = 0; k0 < GIN_D3; k0 += 4) {
    int ka = k0 + 2 * half;
    v2f a;
    a.x = hrow[ka];
    a.y = hrow[ka + 1];
    const float* wr0 = W1 + (size_t)ka * GIN_D3 + l;
    const float* wr1 = wr0 + GIN_D3;
#pragma unroll
    for (int t = 0; t < 12; ++t) {
      v2f b;
      b.x = wr0[t * 16];
      b.y = wr1[t * 16];
      acc[t] = __builtin_amdgcn_wmma_f32_16x16x4_f32(
          false, a, false, b, (short)0, acc[t], false, false);
    }
  }

  float part[8];
#pragma unroll
  for (int v = 0; v < 8; ++v) part[v] = 0.0f;

#pragma unroll
  for (int t = 0; t < 12; ++t) {
    int col = t * 16 + l;
    float bi  = b1[col];
    float w2c = w2[col];
#pragma unroll
    for (int v = 0; v < 8; ++v) {
      float val = fmaxf(acc[t][v] + bi, 0.0f);
      part[v] = fmaf(val, w2c, part[v]);
    }
  }

  // reduce each row's partial across the 16 lanes of its N-group
#pragma unroll
  for (int v = 0; v < 8; ++v) {
    part[v] += __shfl_xor(part[v], 8, 16);
    part[v] += __shfl_xor(part[v], 4, 16);
    part[v] += __shfl_xor(part[v], 2, 16);
    part[v] += __shfl_xor(part[v], 1, 16);
  }

  if (l == 0) {
    float bb = b2[0];
#pragma unroll
    for (int v = 0; v < 8; ++v) {
      int row = row0 + v + 8 * half;
      if (row < rows) out[row] = part[v] + bb;
    }
  }
}

// ---------------------------------------------------------------------------
// launch
// ---------------------------------------------------------------------------
extern "C" void kernel_launch(void* const* d_in, const int* in_sizes, int n_in,
                              void* d_out, int out_size, void* d_ws, size_t ws_size,
                              hipStream_t stream) {
  const float*     x  = (const float*)d_in[0];
  const long long* ei = (const long long*)d_in[1];
  int n_nodes = in_sizes[0] / GIN_D_IN;
  int n_edges = in_sizes[1] / 2;
  const long long* src = ei;
  const long long* dst = ei + n_edges;

  // params: per layer i, base = 2 + 8*i: w1,b1,g,be,rm,rv,w2,b2
  const float* w1[3]; const float* b1[3]; const float* g[3]; const float* be[3];
  const float* rm[3]; const float* rv[3]; const float* w2[3]; const float* b2[3];
  for (int i = 0; i < 3; ++i) {
    int base = 2 + 8 * i;
    w1[i] = (const float*)d_in[base + 0];
    b1[i] = (const float*)d_in[base + 1];
    g[i]  = (const float*)d_in[base + 2];
    be[i] = (const float*)d_in[base + 3];
    rm[i] = (const float*)d_in[base + 4];
    rv[i] = (const float*)d_in[base + 5];
    w2[i] = (const float*)d_in[base + 6];
    b2[i] = (const float*)d_in[base + 7];
  }
  const float* w_lin1 = (const float*)d_in[26];
  const float* b_lin1 = (const float*)d_in[27];
  const float* w_lin2 = (const float*)d_in[28];
  const float* b_lin2 = (const float*)d_in[29];

  float* ws  = (float*)d_ws;
  float* agg = ws;                                   // n_nodes * 68
  float* tmp = agg + (size_t)n_nodes * GIN_D_IN;     // n_nodes * 64
  float* H   = tmp + (size_t)n_nodes * GIN_D_H;      // n_nodes * 192 (concat)

  int m_tiles     = (n_nodes + 15) / 16;
  int gemm_blocks = (m_tiles + 7) / 8;               // 8 waves / 256-thread block

  for (int layer = 0; layer < 3; ++layer) {
    const float* hin;
    int hstride, cols;
    if (layer == 0) { hin = x;                        hstride = GIN_D_IN; cols = GIN_D_IN; }
    else            { hin = H + (layer - 1) * GIN_D_H; hstride = GIN_D3;  cols = GIN_D_H; }
    int cols4 = cols / 4;

    // agg = hin (self term, eps = 0)
    {
      int total  = n_nodes * cols4;
      int blocks = (total + 255) / 256;
      gin_copy_rows<<<blocks, 256, 0, stream>>>(agg, cols, hin, hstride, n_nodes, cols4);
    }
    // agg[dst] += hin[src] over all edges
    {
      int total  = n_edges * cols4;
      int blocks = (total + 255) / 256;
      gin_scatter_add<<<blocks, 256, 0, stream>>>(agg, cols, hin, hstride,
                                                  src, dst, n_edges, cols4);
    }
    // tmp = relu(bn(agg @ w1 + b1))
    gin_mlp_gemm<<<gemm_blocks, 256, 0, stream>>>(
        agg, cols, cols, w1[layer], b1[layer],
        g[layer], be[layer], rm[layer], rv[layer],
        tmp, GIN_D_H, n_nodes, m_tiles);
    // H[:, layer*64 : layer*64+64] = relu(tmp @ w2 + b2)
    gin_mlp_gemm<<<gemm_blocks, 256, 0, stream>>>(
        tmp, GIN_D_H, GIN_D_H, w2[layer], b2[layer],
        nullptr, nullptr, nullptr, nullptr,
        H + layer * GIN_D_H, GIN_D3, n_nodes, m_tiles);
  }

  // out = relu(H @ w_lin1 + b_lin1) @ w_lin2 + b_lin2
  gin_final_head<<<gemm_blocks, 256, 0, stream>>>(
      H, w_lin1, b_lin1, w_lin2, b_lin2, (float*)d_out, n_nodes, m_tiles);

  (void)n_in; (void)out_size; (void)ws_size;
}